// NTXentLoss_24154896073485
// MI455X (gfx1250) — compile-verified
//
#include <hip/hip_runtime.h>
#include <hip/hip_bf16.h>

#define N_EMB   100000
#define D_DIM   128
#define P_PAIRS 8192
#define NEG_CNT 256
#define TEMP_INV (1.0f / 0.07f)

typedef float v2f __attribute__((ext_vector_type(2)));
typedef float v8f __attribute__((ext_vector_type(8)));

// ---------------------------------------------------------------------------
// Kernel 1: emb = X @ W^T + b   using V_WMMA_F32_16X16X4_F32
// One wave computes a 16-row x 128-col tile of emb. K = 128 in steps of 4.
// Fragment layouts per CDNA5 ISA 7.12.2:
//   A (16x4 f32, 2 VGPR):  lane%16 = M row; half-wave selects K pair (0,1)/(2,3)
//   B (4x16 f32, 2 VGPR):  lane%16 = N col; half-wave selects K pair
//   C/D (16x16 f32, 8 VGPR): vgpr r -> M = r + 8*(lane/16), N = lane%16
// Since W is stored row-major [j][k] and B[k][j] = W[j][k], the B fragment
// load has the same (row = lane%16, k = k0 + 2*(lane/16)) pattern as A.
// ---------------------------------------------------------------------------
__global__ void __launch_bounds__(128)
emb_gemm_wmma_kernel(const float* __restrict__ X,
                     const float* __restrict__ W,
                     const float* __restrict__ bias,
                     float* __restrict__ emb,
                     int n_tiles) {
    const int wave = threadIdx.x >> 5;
    const int lane = threadIdx.x & 31;
    const int tile = blockIdx.x * (blockDim.x >> 5) + wave;
    if (tile >= n_tiles) return;

    const int r0    = tile * 16;
    const int mrow  = lane & 15;          // M (for A) / N (for B)
    const int kpair = (lane >> 4) << 1;   // 0 or 2

    v8f acc[8];
#pragma unroll
    for (int t = 0; t < 8; ++t) acc[t] = (v8f){0, 0, 0, 0, 0, 0, 0, 0};

    for (int k0 = 0; k0 < D_DIM; k0 += 4) {
        const float* ap = X + (size_t)(r0 + mrow) * D_DIM + k0 + kpair;
        v2f a;
        a.x = ap[0];
        a.y = ap[1];
#pragma unroll
        for (int t = 0; t < 8; ++t) {
            const int n0 = t * 16;
            const float* bp = W + (size_t)(n0 + mrow) * D_DIM + k0 + kpair;
            v2f b;
            b.x = bp[0];
            b.y = bp[1];
            acc[t] = __builtin_amdgcn_wmma_f32_16x16x4_f32(
                /*neg_a=*/false, a, /*neg_b=*/false, b,
                /*c_mod=*/(short)0, acc[t], /*reuse_a=*/false, /*reuse_b=*/false);
        }
    }

    // Store D with bias added.
#pragma unroll
    for (int t = 0; t < 8; ++t) {
        const int col = t * 16 + (lane & 15);
        const float bv = bias[col];
        const int mbase = (lane >> 4) << 3;   // 0 or 8
#pragma unroll
        for (int r = 0; r < 8; ++r) {
            emb[(size_t)(r0 + mbase + r) * D_DIM + col] = acc[t][r] + bv;
        }
    }
}

// ---------------------------------------------------------------------------
// Reduction helpers (wave32)
// ---------------------------------------------------------------------------
__device__ __forceinline__ float wave_sum(float v) {
#pragma unroll
    for (int off = 16; off > 0; off >>= 1) v += __shfl_xor(v, off, 32);
    return v;
}
__device__ __forceinline__ float wave_max(float v) {
#pragma unroll
    for (int off = 16; off > 0; off >>= 1) v = fmaxf(v, __shfl_xor(v, off, 32));
    return v;
}

// ---------------------------------------------------------------------------
// Kernel 2: per-pair NT-Xent loss. One 256-thread block per pair.
// l_pos = dot(q,k); l_neg[n] = dot(q, emb[neg_idx[p][n]]); stable logsumexp.
// 4 lanes cooperate on each neg row (each lane covers 32 contiguous floats as
// 8 float4 loads -> coalesced 64B segments), partials combined via shfl_xor.
// losses[p] written to workspace (no float atomics -> deterministic).
// ---------------------------------------------------------------------------
__global__ void __launch_bounds__(256)
ntxent_pair_kernel(const float* __restrict__ emb,
                   const int* __restrict__ anchor,
                   const int* __restrict__ partner,
                   const int* __restrict__ negidx,
                   float* __restrict__ losses) {
    const int p   = blockIdx.x;
    const int tid = threadIdx.x;
    const int w   = tid >> 5;
    const int lane = tid & 31;

    __shared__ __align__(16) float q[D_DIM];
    __shared__ float lneg[NEG_CNT];
    __shared__ float red[8];
    __shared__ float s_lpos;

    const int ai = anchor[p];
    const int bi = partner[p];
    const float* qrow = emb + (size_t)ai * D_DIM;
    const float* krow = emb + (size_t)bi * D_DIM;

    // Load q into LDS; partial products for l_pos on threads 0..127.
    float lp = 0.0f;
    if (tid < D_DIM) {
        const float qv = qrow[tid];
        q[tid] = qv;
        lp = qv * krow[tid];
    }
    lp = wave_sum(lp);
    if (lane == 0) red[w] = (w < 4) ? lp : 0.0f;
    __syncthreads();
    if (tid == 0) s_lpos = red[0] + red[1] + red[2] + red[3];
    __syncthreads();

    const int* negp = negidx + (size_t)p * NEG_CNT;
    const int sub = lane & 3;             // which quarter of the row
    const float4* q4 = (const float4*)q;

    // 8 waves x 8 rows/wave = 64 rows per iteration; 4 iterations.
#pragma unroll
    for (int it = 0; it < 4; ++it) {
        const int n = it * 64 + w * 8 + (lane >> 2);
        const int idx = negp[n];
        // Prefetch next iteration's row (gfx1250 global_prefetch_b8).
        const int npf = n + 64;
        if (npf < NEG_CNT) {
            const int pidx = negp[npf];
            __builtin_prefetch(emb + (size_t)pidx * D_DIM + sub * 32, 0, 1);
        }
        const float4* nr = (const float4*)(emb + (size_t)idx * D_DIM);
        float part = 0.0f;
#pragma unroll
        for (int j = 0; j < 8; ++j) {
            const float4 nv = nr[sub * 8 + j];
            const float4 qv = q4[sub * 8 + j];
            part = fmaf(nv.x, qv.x, part);
            part = fmaf(nv.y, qv.y, part);
            part = fmaf(nv.z, qv.z, part);
            part = fmaf(nv.w, qv.w, part);
        }
        // combine the 4 cooperating lanes
        part += __shfl_xor(part, 1, 32);
        part += __shfl_xor(part, 2, 32);
        if (sub == 0) lneg[n] = part;
    }
    __syncthreads();

    // Stable logsumexp over 257 logits (scaled by 1/TEMP).
    const float lpos_logit = s_lpos * TEMP_INV;
    const float my_logit = lneg[tid] * TEMP_INV;   // one neg logit per thread

    // block max
    float m = wave_max((tid == 0) ? fmaxf(my_logit, lpos_logit) : my_logit);
    if (lane == 0) red[w] = m;
    __syncthreads();
    m = red[0];
#pragma unroll
    for (int i = 1; i < 8; ++i) m = fmaxf(m, red[i]);
    __syncthreads();

    // block sum of exp
    float e = __expf(my_logit - m);
    if (tid == 0) e += __expf(lpos_logit - m);
    e = wave_sum(e);
    if (lane == 0) red[w] = e;
    __syncthreads();
    if (tid == 0) {
        float s = red[0];
#pragma unroll
        for (int i = 1; i < 8; ++i) s += red[i];
        losses[p] = (__logf(s) + m) - lpos_logit;
    }
}

// ---------------------------------------------------------------------------
// Kernel 3: out = mean(losses[0:P]) + 0.1 * mean(losses[P:2P]); deterministic.
// ---------------------------------------------------------------------------
__global__ void __launch_bounds__(256)
final_reduce_kernel(const float* __restrict__ losses, float* __restrict__ out) {
    const int tid = threadIdx.x;
    const int w = tid >> 5;
    const int lane = tid & 31;
    __shared__ float red[8];

    float acc = 0.0f;
    for (int i = tid; i < P_PAIRS; i += 256)
        acc += losses[i] + 0.1f * losses[P_PAIRS + i];
    acc = wave_sum(acc);
    if (lane == 0) red[w] = acc;
    __syncthreads();
    if (tid == 0) {
        float s = red[0];
#pragma unroll
        for (int i = 1; i < 8; ++i) s += red[i];
        out[0] = s / (float)P_PAIRS;
    }
}

// ---------------------------------------------------------------------------
extern "C" void kernel_launch(void* const* d_in, const int* in_sizes, int n_in,
                              void* d_out, int out_size, void* d_ws, size_t ws_size,
                              hipStream_t stream) {
    const float* X    = (const float*)d_in[0];   // embeddings [N,D]
    const float* W    = (const float*)d_in[1];   // W [D,D]
    const float* bias = (const float*)d_in[2];   // b [D]
    const int* pa  = (const int*)d_in[3];        // pos_anchor [P]
    const int* pp  = (const int*)d_in[4];        // pos_partner [P]
    const int* ni  = (const int*)d_in[5];        // neg_idx [P,NEG]
    const int* wa  = (const int*)d_in[6];        // weak_anchor [P]
    const int* wp  = (const int*)d_in[7];        // weak_partner [P]
    const int* wni = (const int*)d_in[8];        // weak_neg_idx [P,NEG]
    float* out = (float*)d_out;

    float* emb    = (float*)d_ws;                         // N*D floats (51.2 MB)
    float* losses = emb + (size_t)N_EMB * D_DIM;          // 2*P floats

    // 1) GEMM: emb = X @ W^T + b   (WMMA f32)
    const int n_tiles = N_EMB / 16;                       // 6250
    const int waves_per_block = 4;
    dim3 gblk(128);
    dim3 ggrd((n_tiles + waves_per_block - 1) / waves_per_block);
    emb_gemm_wmma_kernel<<<ggrd, gblk, 0, stream>>>(X, W, bias, emb, n_tiles);

    // 2) Per-pair losses, both branches (gathers hit L2: emb=51MB < 192MB L2)
    ntxent_pair_kernel<<<P_PAIRS, 256, 0, stream>>>(emb, pa, pp, ni, losses);
    ntxent_pair_kernel<<<P_PAIRS, 256, 0, stream>>>(emb, wa, wp, wni,
                                                    losses + P_PAIRS);

    // 3) Deterministic final reduction
    final_reduce_kernel<<<1, 256, 0, stream>>>(losses, out);
}